// CustomAttention_10342281249213
// MI455X (gfx1250) — compile-verified
//
#include <hip/hip_runtime.h>
#include <hip/hip_bf16.h>

typedef float v2f __attribute__((ext_vector_type(2)));
typedef float v4f __attribute__((ext_vector_type(4)));
typedef float v8f __attribute__((ext_vector_type(8)));

#define NDIM  1024
#define BATCH 64
#define LSEQ  2048

// Hardware TANH (CDNA5 TRANS unit) with safe fallback.
__device__ __forceinline__ float htanh(float x) {
#if __has_builtin(__builtin_amdgcn_tanhf)
  return __builtin_amdgcn_tanhf(x);
#else
  return tanhf(x);
#endif
}

// ---------------------------------------------------------------------------
// Kernel 1: decoder projection  df[m,n] = sum_k ds[m,k]*W[n,k] + b[n]
// One wave32 per 16x16 output tile, V_WMMA_F32_16X16X4_F32, K-loop of 256.
// A (16x4 f32): lane<16 -> M=lane, K={0,1}; lane>=16 -> M=lane-16, K={2,3}
// B (4x16 f32): lane<16 -> N=lane, K={0,1}; lane>=16 -> N=lane-16, K={2,3}
// ---------------------------------------------------------------------------
__global__ __launch_bounds__(32) void k_proj(const float* __restrict__ ds,
                                             const float* __restrict__ W,
                                             const float* __restrict__ bias,
                                             float* __restrict__ df) {
  int tile = blockIdx.x;                // 256 tiles = 4 (M) x 64 (N)
  int m0 = (tile & 3) * 16;
  int n0 = (tile >> 2) * 16;
  int lane = threadIdx.x;
  int half = lane >> 4;
  int idx  = lane & 15;

  const float* Arow = ds + (size_t)(m0 + idx) * NDIM + 2 * half;
  const float* Brow = W  + (size_t)(n0 + idx) * NDIM + 2 * half;

  v8f acc = {};
  for (int k = 0; k < NDIM; k += 4) {
    v2f a = *(const v2f*)(Arow + k);
    v2f b = *(const v2f*)(Brow + k);
    acc = __builtin_amdgcn_wmma_f32_16x16x4_f32(false, a, false, b,
                                                (short)0, acc, false, false);
  }

  float bn = bias[n0 + idx];
  int mbase = m0 + 8 * half;
#pragma unroll
  for (int j = 0; j < 8; ++j)
    df[(size_t)(mbase + j) * NDIM + (n0 + idx)] = acc[j] + bn;
}

// ---------------------------------------------------------------------------
// Kernel 2: scores[b,l] = v . tanh(ef[b,l,:] + df[b,:])
// df row + v staged in LDS as float4; one wave per l; per-lane b128
// non-temporal streaming loads of ef (512B per wave per instruction).
// ---------------------------------------------------------------------------
#define L_PER_BLOCK 64
__global__ __launch_bounds__(256) void k_scores(const float* __restrict__ ef,
                                                const float* __restrict__ df,
                                                const float* __restrict__ v,
                                                float* __restrict__ scores) {
  __shared__ v4f s_df[NDIM / 4];   // 256 float4
  __shared__ v4f s_v[NDIM / 4];
  int b  = blockIdx.y;
  int l0 = blockIdx.x * L_PER_BLOCK;

  {
    int i = threadIdx.x;           // NDIM/4 == blockDim.x == 256
    s_df[i] = ((const v4f*)(df + (size_t)b * NDIM))[i];
    s_v[i]  = ((const v4f*)v)[i];
  }
  __syncthreads();

  int wave = threadIdx.x >> 5;
  int lane = threadIdx.x & 31;
  for (int li = wave; li < L_PER_BLOCK; li += 8) {
    int l = l0 + li;
    const v4f* row = (const v4f*)(ef + ((size_t)b * LSEQ + l) * NDIM);
    float sum = 0.f;
#pragma unroll
    for (int i = 0; i < NDIM / 128; ++i) {     // 8 iterations
      int n4 = lane + 32 * i;                  // coalesced float4 index
      v4f e  = __builtin_nontemporal_load(row + n4);
      v4f d  = s_df[n4];
      v4f vv = s_v[n4];
      sum += vv.x * htanh(e.x + d.x);
      sum += vv.y * htanh(e.y + d.y);
      sum += vv.z * htanh(e.z + d.z);
      sum += vv.w * htanh(e.w + d.w);
    }
    for (int off = 16; off > 0; off >>= 1)
      sum += __shfl_xor(sum, off, 32);
    if (lane == 0) scores[(size_t)b * LSEQ + l] = sum;
  }
}

// ---------------------------------------------------------------------------
// Kernel 3: masked softmax with renormalization; attn -> d_out slot.
// attn = exp(s - max) * mask / sum(exp(s - max) * mask)   (== reference)
// ---------------------------------------------------------------------------
__global__ __launch_bounds__(256) void k_softmax(const float* __restrict__ scores,
                                                 const int* __restrict__ mask,
                                                 float* __restrict__ attn) {
  __shared__ float redmax[8];
  __shared__ float redsum[8];
  int b = blockIdx.x;
  int t = threadIdx.x;
  const float* srow = scores + (size_t)b * LSEQ;
  const int*   mrow = mask   + (size_t)b * LSEQ;
  float*       arow = attn   + (size_t)b * LSEQ;

  float m = -1e30f;
  for (int l = t; l < LSEQ; l += 256) m = fmaxf(m, srow[l]);
  for (int off = 16; off > 0; off >>= 1) m = fmaxf(m, __shfl_xor(m, off, 32));
  if ((t & 31) == 0) redmax[t >> 5] = m;
  __syncthreads();
  m = redmax[0];
#pragma unroll
  for (int w = 1; w < 8; ++w) m = fmaxf(m, redmax[w]);

  float s = 0.f;
  for (int l = t; l < LSEQ; l += 256) {
    float e = expf(srow[l] - m) * (float)mrow[l];
    arow[l] = e;
    s += e;
  }
  for (int off = 16; off > 0; off >>= 1) s += __shfl_xor(s, off, 32);
  if ((t & 31) == 0) redsum[t >> 5] = s;
  __syncthreads();
  s = 0.f;
#pragma unroll
  for (int w = 0; w < 8; ++w) s += redsum[w];

  float inv = 1.f / s;
  for (int l = t; l < LSEQ; l += 256) arow[l] *= inv;
}

// ---------------------------------------------------------------------------
// Kernel 4: context partials: part[chunk,b,n] = sum_{l in chunk} attn[b,l]*eo[b,l,n]
// attn chunk in LDS; each thread owns one float4 column group (n = 4t..4t+3),
// per-lane b128 non-temporal streaming loads of eo.
// ---------------------------------------------------------------------------
#define LCHUNK 128
__global__ __launch_bounds__(256) void k_context(const float* __restrict__ eo,
                                                 const float* __restrict__ attn,
                                                 float* __restrict__ part) {
  __shared__ float s_a[LCHUNK];
  int chunk = blockIdx.x;               // 16 chunks of 128
  int b     = blockIdx.y;
  int l0    = chunk * LCHUNK;
  int t     = threadIdx.x;

  for (int i = t; i < LCHUNK; i += 256) s_a[i] = attn[(size_t)b * LSEQ + l0 + i];
  __syncthreads();

  v4f acc = {};
  const v4f* base = (const v4f*)(eo + ((size_t)b * LSEQ + l0) * NDIM);
#pragma unroll 4
  for (int li = 0; li < LCHUNK; ++li) {
    float a = s_a[li];
    v4f r = __builtin_nontemporal_load(base + (size_t)li * (NDIM / 4) + t);
    acc += r * a;
  }
  v4f* p = (v4f*)(part + ((size_t)chunk * BATCH + b) * NDIM);
  p[t] = acc;   // n = 4t .. 4t+3  (RT policy: re-read by k_reduce from L2)
}

// ---------------------------------------------------------------------------
// Kernel 5: reduce 16 partials -> context in d_out (float4 lanes).
// ---------------------------------------------------------------------------
__global__ __launch_bounds__(256) void k_reduce(const float* __restrict__ part,
                                                float* __restrict__ ctx) {
  int i = blockIdx.x * 256 + threadIdx.x;   // over B*N/4 = 16384 float4
  v4f s = {};
#pragma unroll
  for (int c = 0; c < 16; ++c)
    s += ((const v4f*)part)[(size_t)c * (BATCH * NDIM / 4) + i];
  ((v4f*)ctx)[i] = s;
}

// ---------------------------------------------------------------------------
extern "C" void kernel_launch(void* const* d_in, const int* in_sizes, int n_in,
                              void* d_out, int out_size, void* d_ws, size_t ws_size,
                              hipStream_t stream) {
  (void)in_sizes; (void)n_in; (void)out_size; (void)ws_size;
  const float* ds   = (const float*)d_in[0];  // [B, N]
  const float* eo   = (const float*)d_in[1];  // [B, L, N]
  const float* ef   = (const float*)d_in[2];  // [B, L, N]
  const int*   mask = (const int*)  d_in[3];  // [B, L]
  const float* W    = (const float*)d_in[4];  // [N, N]
  const float* bias = (const float*)d_in[5];  // [N]
  const float* v    = (const float*)d_in[6];  // [N]

  float* ctx  = (float*)d_out;                // [B, N]
  float* attn = ctx + (size_t)BATCH * NDIM;   // [B, L]

  float* df     = (float*)d_ws;                       // B*N
  float* scores = df + (size_t)BATCH * NDIM;          // B*L
  float* part   = scores + (size_t)BATCH * LSEQ;      // 16*B*N

  k_proj<<<256, 32, 0, stream>>>(ds, W, bias, df);
  k_scores<<<dim3(LSEQ / L_PER_BLOCK, BATCH), 256, 0, stream>>>(ef, df, v, scores);
  k_softmax<<<BATCH, 256, 0, stream>>>(scores, mask, attn);
  k_context<<<dim3(LSEQ / LCHUNK, BATCH), 256, 0, stream>>>(eo, attn, part);
  k_reduce<<<(BATCH * NDIM) / 1024, 256, 0, stream>>>(part, ctx);
}